// LightGCL_26439818674747
// MI455X (gfx1250) — compile-verified
//
#include <hip/hip_runtime.h>
#include <math.h>

// ---------------- problem constants (match reference) ----------------
#define N_U 100000
#define N_I 50000
#define DIM 64
#define NNZ 1000000
#define NLAYER 2
#define BATCH 2048
#define TEMP_INV 5.0f        // 1/0.2
#define LAMBDA_1 0.2f
#define DROPOUT 0.1f
#define EPS_AUG 0.05f

typedef float v2f __attribute__((ext_vector_type(2)));
typedef float v8f __attribute__((ext_vector_type(8)));

__device__ __forceinline__ float leaky05(float x) { return (x >= 0.0f) ? x : 0.5f * x; }

// =====================================================================
// 1) Sparse message passing: Z[seg[e]] += dv_e * E[gat[e]]  (atomic scatter)
//    one wave per edge, each lane handles 2 of the 64 dims
// =====================================================================
__global__ void k_spmm(const float* __restrict__ adj_vals,
                       const float* __restrict__ drop,
                       const int* __restrict__ seg,
                       const int* __restrict__ gat,
                       const float* __restrict__ E,
                       float* __restrict__ Z) {
    long long g = (long long)blockIdx.x * blockDim.x + threadIdx.x;
    int e = (int)(g >> 5);
    int lane = (int)(g & 31);
    if (e >= NNZ) return;
    float keep = (drop[e] >= DROPOUT) ? 1.0f : 0.0f;
    float dv = adj_vals[e] * keep * (1.0f / (1.0f - DROPOUT));
    if (dv == 0.0f) return;
    const float* src = E + (size_t)gat[e] * DIM;
    float* dst = Z + (size_t)seg[e] * DIM;
    atomicAdd(&dst[lane],      dv * src[lane]);
    atomicAdd(&dst[lane + 32], dv * src[lane + 32]);
}

// Z <- leaky(Z) + Eprev   (in place; Z becomes E_next)
__global__ void k_update_E(float* __restrict__ Z, const float* __restrict__ Eprev, int count) {
    int i = blockIdx.x * blockDim.x + threadIdx.x;
    if (i >= count) return;
    Z[i] = leaky05(Z[i]) + Eprev[i];
}

// =====================================================================
// 2) K-reduction GEMM:  C[64,64] += A[64,N] @ B[N,64]  (A stride = N)
//    split-K: each block reduces a 500-wide K chunk; 8 waves x 2 tiles
//    => 16 WMMA tiles (full 64x64), atomics into C.
// =====================================================================
#define KRED_CHUNK 500
__global__ void k_gemm_kred(const float* __restrict__ A, int strideA,
                            const float* __restrict__ Bm,
                            float* __restrict__ C) {
    int tid = threadIdx.x;
    int wave = tid >> 5, lane = tid & 31;
    int hi = lane >> 4, lo = lane & 15;
    int n0 = blockIdx.x * KRED_CHUNK;
    int t0 = wave * 2;          // tiles t0, t0+1 share the same row-tile
    int mt = t0 >> 2;
    int nt0 = t0 & 3;
    v8f acc0 = {0.f,0.f,0.f,0.f,0.f,0.f,0.f,0.f};
    v8f acc1 = {0.f,0.f,0.f,0.f,0.f,0.f,0.f,0.f};
    const float* Arow = A + (size_t)(mt * 16 + lo) * (size_t)strideA;
    for (int k = n0; k < n0 + KRED_CHUNK; k += 4) {
        int kk = k + 2 * hi;
        v2f a; a.x = Arow[kk]; a.y = Arow[kk + 1];
        v2f b0; b0.x = Bm[(size_t)kk * DIM + nt0 * 16 + lo];
                b0.y = Bm[(size_t)(kk + 1) * DIM + nt0 * 16 + lo];
        v2f b1; b1.x = Bm[(size_t)kk * DIM + (nt0 + 1) * 16 + lo];
                b1.y = Bm[(size_t)(kk + 1) * DIM + (nt0 + 1) * 16 + lo];
        acc0 = __builtin_amdgcn_wmma_f32_16x16x4_f32(false, a, false, b0, (short)0, acc0, false, false);
        acc1 = __builtin_amdgcn_wmma_f32_16x16x4_f32(false, a, false, b1, (short)0, acc1, false, false);
    }
#pragma unroll
    for (int v = 0; v < 8; v++) {
        int m = mt * 16 + v + 8 * hi;
        atomicAdd(&C[(size_t)m * DIM + nt0 * 16 + lo],       acc0[v]);
        atomicAdd(&C[(size_t)m * DIM + (nt0 + 1) * 16 + lo], acc1[v]);
    }
}

// =====================================================================
// 3) Out[M,64] = act( A[M,64] @ B[64,64] ), B staged in LDS.
//    1 wave per 16-row strip; act: 0=identity, 1=leaky(0.5)
// =====================================================================
__global__ void k_gemm_n64(const float* __restrict__ A,
                           const float* __restrict__ Bm,
                           float* __restrict__ Out, int M, int act) {
    __shared__ float Bs[DIM * DIM];
    int tid = threadIdx.x;
    for (int i = tid; i < DIM * DIM; i += blockDim.x) Bs[i] = Bm[i];
    __syncthreads();
    int wave = tid >> 5, lane = tid & 31;
    int hi = lane >> 4, lo = lane & 15;
    int row0 = (blockIdx.x * (blockDim.x >> 5) + wave) * 16;
    if (row0 >= M) return;
    v8f acc[4];
#pragma unroll
    for (int nt = 0; nt < 4; nt++) acc[nt] = (v8f){0.f,0.f,0.f,0.f,0.f,0.f,0.f,0.f};
    const float* Arow = A + (size_t)(row0 + lo) * DIM;
    for (int k0 = 0; k0 < DIM; k0 += 4) {
        int kk = k0 + 2 * hi;
        v2f a; a.x = Arow[kk]; a.y = Arow[kk + 1];
#pragma unroll
        for (int nt = 0; nt < 4; nt++) {
            v2f b; b.x = Bs[kk * DIM + nt * 16 + lo];
                   b.y = Bs[(kk + 1) * DIM + nt * 16 + lo];
            acc[nt] = __builtin_amdgcn_wmma_f32_16x16x4_f32(false, a, false, b, (short)0, acc[nt], false, false);
        }
    }
#pragma unroll
    for (int nt = 0; nt < 4; nt++) {
#pragma unroll
        for (int v = 0; v < 8; v++) {
            float x = acc[nt][v];
            if (act) x = leaky05(x);
            Out[(size_t)(row0 + v + 8 * hi) * DIM + nt * 16 + lo] = x;
        }
    }
}

// =====================================================================
// 4) Gather+perturb+scale:  Out[b] = (svd[id] + sign(svd[id])*l2norm(noise[id])*EPS) * s
//    one wave per batch row, lane -> dims (lane, lane+32)
// =====================================================================
__global__ void k_gather_svd(const float* __restrict__ svd,
                             const float* __restrict__ noise,
                             const float* __restrict__ s,
                             const int* __restrict__ ids,
                             float* __restrict__ Out) {
    int g = blockIdx.x * blockDim.x + threadIdx.x;
    int b = g >> 5, lane = g & 31;
    if (b >= BATCH) return;
    int id = ids[b];
    const float* nrow = noise + (size_t)id * DIM;
    float n0 = nrow[lane], n1 = nrow[lane + 32];
    float ss = n0 * n0 + n1 * n1;
    for (int off = 16; off; off >>= 1) ss += __shfl_xor(ss, off, 32);
    float inv = 1.0f / fmaxf(sqrtf(ss), 1e-12f);
    const float* srow = svd + (size_t)id * DIM;
    float sv0 = srow[lane], sv1 = srow[lane + 32];
    float sg0 = (float)(sv0 > 0.f) - (float)(sv0 < 0.f);
    float sg1 = (float)(sv1 > 0.f) - (float)(sv1 < 0.f);
    Out[(size_t)b * DIM + lane]      = (sv0 + sg0 * n0 * inv * EPS_AUG) * s[lane];
    Out[(size_t)b * DIM + lane + 32] = (sv1 + sg1 * n1 * inv * EPS_AUG) * s[lane + 32];
}

// row-wise l2 normalize [M,64] in place (one wave per row)
__global__ void k_rownorm(float* __restrict__ X, int M) {
    int g = blockIdx.x * blockDim.x + threadIdx.x;
    int r = g >> 5, lane = g & 31;
    if (r >= M) return;
    float* row = X + (size_t)r * DIM;
    float a = row[lane], b = row[lane + 32];
    float ss = a * a + b * b;
    for (int off = 16; off; off >>= 1) ss += __shfl_xor(ss, off, 32);
    float inv = 1.0f / fmaxf(sqrtf(ss), 1e-12f);
    row[lane] = a * inv; row[lane + 32] = b * inv;
}

// =====================================================================
// 5) Similarity tiles: sim = h1 @ h2^T, rowsum += exp(sim/T), posd = exp(diag/T)
//    one wave per 16x16 tile (128x128 tiles); WMMA + in-wave row reduce
// =====================================================================
__global__ void k_sim(const float* __restrict__ H1,
                      const float* __restrict__ H2,
                      float* __restrict__ rowsum,
                      float* __restrict__ posd) {
    int tid = threadIdx.x;
    int wave = tid >> 5, lane = tid & 31;
    int hi = lane >> 4, lo = lane & 15;
    int tile = blockIdx.x * (blockDim.x >> 5) + wave;   // 128*128 tiles
    int bt = tile >> 7;
    int jt = tile & 127;
    int row0 = bt * 16, col0 = jt * 16;
    v8f acc = {0.f,0.f,0.f,0.f,0.f,0.f,0.f,0.f};
    const float* Arow = H1 + (size_t)(row0 + lo) * DIM;
    const float* Brow = H2 + (size_t)(col0 + lo) * DIM;   // B[k][n] = H2[n][k]
    for (int k0 = 0; k0 < DIM; k0 += 4) {
        int kk = k0 + 2 * hi;
        v2f a; a.x = Arow[kk]; a.y = Arow[kk + 1];
        v2f b; b.x = Brow[kk]; b.y = Brow[kk + 1];
        acc = __builtin_amdgcn_wmma_f32_16x16x4_f32(false, a, false, b, (short)0, acc, false, false);
    }
#pragma unroll
    for (int v = 0; v < 8; v++) {
        int m = v + 8 * hi;
        float e = expf(acc[v] * TEMP_INV);
        if (bt == jt && lo == m) posd[row0 + m] = e;   // unique writer per row
        float r = e;
        for (int off = 1; off < 16; off <<= 1) r += __shfl_xor(r, off, 32);
        if (lo == 0) atomicAdd(&rowsum[row0 + m], r);
    }
}

// InfoNCE finalize: loss_s += sum_b -log(p_b/(S_b+1e-8)+1e-8)*mask_b
__global__ void k_cl_fin(const float* __restrict__ rowsum,
                         const float* __restrict__ posd,
                         const float* __restrict__ mask_raw,
                         float* __restrict__ loss_s) {
    int t = blockIdx.x * blockDim.x + threadIdx.x;
    if (t >= BATCH) return;
    float m = (mask_raw[t] > 0.5f) ? 1.0f : 0.0f;
    float val = -logf(posd[t] / (rowsum[t] + 1e-8f) + 1e-8f) * m;
    for (int off = 16; off; off >>= 1) val += __shfl_xor(val, off, 32);
    if ((t & 31) == 0) atomicAdd(loss_s, val);
}

// =====================================================================
// 6) BPR hinge loss; E-sums formed on the fly from the 3 per-layer matrices
// =====================================================================
__global__ void k_bpr(const float* __restrict__ Eu0, const float* __restrict__ Eu1,
                      const float* __restrict__ Eu2, const float* __restrict__ Ei0,
                      const float* __restrict__ Ei1, const float* __restrict__ Ei2,
                      const int* __restrict__ uids, const int* __restrict__ pos,
                      const int* __restrict__ neg, float* __restrict__ loss_r) {
    int g = blockIdx.x * blockDim.x + threadIdx.x;
    int b = g >> 5, lane = g & 31;
    if (b >= BATCH) return;
    size_t u = (size_t)uids[b] * DIM, p = (size_t)pos[b] * DIM, n = (size_t)neg[b] * DIM;
    float ps = 0.f, ns = 0.f;
#pragma unroll
    for (int k = 0; k < 2; k++) {
        int d = lane + k * 32;
        float ue = Eu0[u + d] + Eu1[u + d] + Eu2[u + d];
        float pe = Ei0[p + d] + Ei1[p + d] + Ei2[p + d];
        float ne = Ei0[n + d] + Ei1[n + d] + Ei2[n + d];
        ps += ue * pe; ns += ue * ne;
    }
    for (int off = 16; off; off >>= 1) {
        ps += __shfl_xor(ps, off, 32);
        ns += __shfl_xor(ns, off, 32);
    }
    if (lane == 0)
        atomicAdd(loss_r, fmaxf(0.0f, 1.0f - ps + ns) * (1.0f / (float)BATCH));
}

__global__ void k_combine(const float* __restrict__ acc, float* __restrict__ out) {
    float lr = acc[0], ls = acc[1];
    out[0] = lr + LAMBDA_1 * ls;
    out[1] = lr;
    out[2] = ls;
}

// =====================================================================
// host orchestration
// =====================================================================
extern "C" void kernel_launch(void* const* d_in, const int* in_sizes, int n_in,
                              void* d_out, int out_size, void* d_ws, size_t ws_size,
                              hipStream_t stream) {
    (void)in_sizes; (void)n_in; (void)out_size; (void)ws_size;
    const float* E_u_0 = (const float*)d_in[0];
    const float* E_i_0 = (const float*)d_in[1];
    const float* svd_u = (const float*)d_in[2];
    const float* svd_v = (const float*)d_in[3];
    const float* s_vec = (const float*)d_in[4];
    const float* ut    = (const float*)d_in[5];
    const float* vt    = (const float*)d_in[6];
    const float* W_u   = (const float*)d_in[7];
    const float* W_i   = (const float*)d_in[8];
    const float* adj_vals  = (const float*)d_in[9];
    const float* drop_rand = (const float*)d_in[10];
    const float* noise_u1 = (const float*)d_in[11];
    const float* noise_v1 = (const float*)d_in[12];
    const float* noise_u2 = (const float*)d_in[13];
    const float* noise_v2 = (const float*)d_in[14];
    const float* u_mask = (const float*)d_in[15];
    const float* i_mask = (const float*)d_in[16];
    const int* adj_rows = (const int*)d_in[17];
    const int* adj_cols = (const int*)d_in[18];
    const int* uids = (const int*)d_in[19];
    const int* iids = (const int*)d_in[20];
    const int* pos  = (const int*)d_in[21];
    const int* neg  = (const int*)d_in[22];
    float* out = (float*)d_out;

    // workspace carve-up (floats)
    const size_t NU64 = (size_t)N_U * DIM, NI64 = (size_t)N_I * DIM, B64 = (size_t)BATCH * DIM;
    float* ws = (float*)d_ws;
    float* Eu1  = ws;               // layer-1 user emb (also Z accumulator)
    float* Eu2  = Eu1 + NU64;       // layer-2 user emb
    float* Ei1  = Eu2 + NU64;
    float* Ei2  = Ei1 + NI64;
    float* vtEi = Ei2 + NI64;       // 64x64
    float* utEu = vtEi + DIM * DIM; // 64x64
    float* Us1  = utEu + DIM * DIM; // gathered perturbed*scaled rows, view 1
    float* Us2  = Us1 + B64;
    float* G1   = Us2 + B64;
    float* G2   = G1 + B64;
    float* H1   = G2 + B64;
    float* H2   = H1 + B64;
    float* rowsum = H2 + B64;       // [BATCH]
    float* posd   = rowsum + BATCH; // [BATCH]
    float* lossacc = posd + BATCH;  // [0]=loss_r, [1]=loss_s

    hipMemsetAsync(lossacc, 0, 2 * sizeof(float), stream);

    const int TB = 256;
    const int grid_spmm = (int)(((long long)NNZ * 32 + TB - 1) / TB);
    const int grid_wave_batch = (BATCH * 32) / TB;         // 256
    const int grid_gemm_b = (BATCH + 127) / 128;           // 16 (8 waves * 16 rows)
    const int grid_sim = (128 * 128) / 8;                  // 2048 blocks, 8 tiles/block

    const float* Eu_prev = E_u_0;
    const float* Ei_prev = E_i_0;

    for (int l = 1; l <= NLAYER; l++) {
        float* Zu = (l == 1) ? Eu1 : Eu2;
        float* Zi = (l == 1) ? Ei1 : Ei2;
        hipMemsetAsync(Zu, 0, NU64 * sizeof(float), stream);
        hipMemsetAsync(Zi, 0, NI64 * sizeof(float), stream);
        hipMemsetAsync(vtEi, 0, DIM * DIM * sizeof(float), stream);
        hipMemsetAsync(utEu, 0, DIM * DIM * sizeof(float), stream);

        const float* drop_u = drop_rand + ((size_t)(l - 1) * 2 + 0) * NNZ;
        const float* drop_i = drop_rand + ((size_t)(l - 1) * 2 + 1) * NNZ;
        k_spmm<<<grid_spmm, TB, 0, stream>>>(adj_vals, drop_u, adj_rows, adj_cols, Ei_prev, Zu);
        k_spmm<<<grid_spmm, TB, 0, stream>>>(adj_vals, drop_i, adj_cols, adj_rows, Eu_prev, Zi);

        k_gemm_kred<<<N_I / KRED_CHUNK, TB, 0, stream>>>(vt, N_I, Ei_prev, vtEi);
        k_gemm_kred<<<N_U / KRED_CHUNK, TB, 0, stream>>>(ut, N_U, Eu_prev, utEu);

        k_update_E<<<(int)((NU64 + TB - 1) / TB), TB, 0, stream>>>(Zu, Eu_prev, (int)NU64);
        k_update_E<<<(int)((NI64 + TB - 1) / TB), TB, 0, stream>>>(Zi, Ei_prev, (int)NI64);

        // ---- contrastive losses: user side then item side ----
        for (int side = 0; side < 2; side++) {
            const float* svd  = side ? svd_v : svd_u;
            const float* nz1  = side ? noise_v1 : noise_u1;
            const float* nz2  = side ? noise_v2 : noise_u2;
            const int*   ids  = side ? iids : uids;
            const float* proj = side ? utEu : vtEi;
            const float* W    = (side ? W_i : W_u) + (size_t)(l - 1) * DIM * DIM;
            const float* msk  = (side ? i_mask : u_mask) + (size_t)(l - 1) * BATCH;

            k_gather_svd<<<grid_wave_batch, TB, 0, stream>>>(svd, nz1, s_vec, ids, Us1);
            k_gather_svd<<<grid_wave_batch, TB, 0, stream>>>(svd, nz2, s_vec, ids, Us2);
            k_gemm_n64<<<grid_gemm_b, TB, 0, stream>>>(Us1, proj, G1, BATCH, 1);
            k_gemm_n64<<<grid_gemm_b, TB, 0, stream>>>(Us2, proj, G2, BATCH, 1);
            k_rownorm<<<grid_wave_batch, TB, 0, stream>>>(G1, BATCH);
            k_rownorm<<<grid_wave_batch, TB, 0, stream>>>(G2, BATCH);
            k_gemm_n64<<<grid_gemm_b, TB, 0, stream>>>(G1, W, H1, BATCH, 0);
            k_gemm_n64<<<grid_gemm_b, TB, 0, stream>>>(G2, W, H2, BATCH, 0);
            hipMemsetAsync(rowsum, 0, BATCH * sizeof(float), stream);
            k_sim<<<grid_sim, TB, 0, stream>>>(H1, H2, rowsum, posd);
            k_cl_fin<<<BATCH / TB, TB, 0, stream>>>(rowsum, posd, msk, lossacc + 1);
        }

        Eu_prev = Zu;   // layer l user embedding
        Ei_prev = Zi;
    }

    k_bpr<<<grid_wave_batch, TB, 0, stream>>>(E_u_0, Eu1, Eu2, E_i_0, Ei1, Ei2,
                                              uids, pos, neg, lossacc);
    k_combine<<<1, 1, 0, stream>>>(lossacc, out);
}